// Decoder_45784351375646
// MI455X (gfx1250) — compile-verified
//
#include <hip/hip_runtime.h>
#include <hip/hip_bf16.h>

// MI455X / gfx1250 decoder scan.
// Key insight: the 63-step scan is fully batch-independent -> one persistent
// kernel, 128 workgroups x 16 batch rows, workgroup-local barriers only.
// GEMMs use V_WMMA_F32_16X16X4_F32 (f32-exact). Weight B-fragments are
// register-resident across all 63 steps. u_out precomputed into d_ws (32 MB).

typedef float v2f __attribute__((ext_vector_type(2)));
typedef float v8f __attribute__((ext_vector_type(8)));

#define WIN   64
#define BATCH 2048
#define E_    64
#define D_    64
#define G_    (4 * D_)
#define BT    16
#define NSTEP (WIN - 1)

// ---------------------------------------------------------------------------
// Kernel A: u_out[w,b,:] = h_enc[w,b,:] @ u_w^T + u_b     ((W*B) x E)
// ---------------------------------------------------------------------------
__global__ __launch_bounds__(256) void u_out_kernel(
    const float* __restrict__ h_enc, const float* __restrict__ u_w,
    const float* __restrict__ u_b, float* __restrict__ u_out) {
  __shared__ float rowbuf[4][E_];
  const int t = threadIdx.x;
  const size_t row0 = (size_t)blockIdx.x * 4;
  rowbuf[t >> 6][t & 63] = h_enc[row0 * E_ + t];  // 4 rows = 256 floats
  __syncthreads();
  const int r = t >> 6, e = t & 63;
  float acc = u_b[e];
#pragma unroll 8
  for (int k = 0; k < E_; ++k) acc += rowbuf[r][k] * u_w[e * E_ + k];
  u_out[(row0 + r) * E_ + e] = acc;
}

// ---------------------------------------------------------------------------
// Kernel B: persistent per-batch-tile decoder scan (63 steps)
// ---------------------------------------------------------------------------
__global__ __launch_bounds__(256) void decoder_kernel(
    const float* __restrict__ h_enc, const float* __restrict__ y,
    const float* __restrict__ w_ih, const float* __restrict__ w_hh,
    const float* __restrict__ b_ih, const float* __restrict__ b_hh,
    const float* __restrict__ v_w, const float* __restrict__ w_w,
    const float* __restrict__ w_b, const float* __restrict__ p_w,
    const float* __restrict__ p_b, const float* __restrict__ y_w,
    const float* __restrict__ y_b, const float* __restrict__ u_out,
    float* __restrict__ out) {
  __shared__ float s_h[BT][D_ + 1];
  __shared__ float s_c[BT][D_ + 1];
  __shared__ float s_wout[BT][E_ + 1];
  __shared__ float s_score[WIN][BT + 1];
  __shared__ float s_ctx[BT][E_ + 1];
  __shared__ float s_gates[BT][G_ + 1];
  __shared__ float s_vw[E_], s_wb[E_], s_bias[G_], s_wih[G_];
  __shared__ float s_pw[D_ + 1], s_yw[2 * D_], s_ytil[BT];

  const int t     = threadIdx.x;
  const int lane  = t & 31;
  const int wave  = t >> 5;
  const int b0    = blockIdx.x * BT;
  const int arow  = lane & 15;         // A-matrix row (ISA 7.12.2, 32-bit A 16x4)
  const int kbase = (lane >> 4) * 2;   // lanes 0-15: K=0,1 ; lanes 16-31: K=2,3
  const int ccol  = lane & 15;         // C/D: N = lane % 16
  const int crow0 = (lane >> 4) * 8;   // C/D: VGPR i -> M = i (+8 for hi lanes)

  // ---- small weights to LDS; zero h,c ----
  if (t < E_)     { s_vw[t] = v_w[t]; s_wb[t] = w_b[t]; }
  if (t < G_)     { s_bias[t] = b_ih[t] + b_hh[t]; s_wih[t] = w_ih[t]; }
  if (t < D_ + 1) s_pw[t] = p_w[t];
  if (t < 2 * D_) s_yw[t] = y_w[t];
  for (int o = t; o < BT * (D_ + 1); o += 256) {
    (&s_h[0][0])[o] = 0.f;
    (&s_c[0][0])[o] = 0.f;
  }
  const float pb = p_b[0];

  // ---- loop-invariant B fragments -> registers ----
  // gates GEMM: B[k][n] = w_hh[n][k], each wave owns N-tiles 2w, 2w+1
  v2f Bg[2][16];
#pragma unroll
  for (int rep = 0; rep < 2; ++rep) {
    const int n = (wave * 2 + rep) * 16 + ccol;
#pragma unroll
    for (int kk = 0; kk < 16; ++kk) {
      const int k = kk * 4 + kbase;
      Bg[rep][kk].x = w_hh[n * D_ + k];
      Bg[rep][kk].y = w_hh[n * D_ + k + 1];
    }
  }
  // w_out GEMM: B[k][n] = w_w[n][k] (K=128), waves 0..3 own N-tiles 0..3
  v2f Bw[32] = {};
  if (wave < 4) {
    const int n = wave * 16 + ccol;
#pragma unroll
    for (int kk = 0; kk < 32; ++kk) {
      const int k = kk * 4 + kbase;
      Bw[kk].x = w_w[n * (2 * D_) + k];
      Bw[kk].y = w_w[n * (2 * D_) + k + 1];
    }
  }
  __syncthreads();

  for (int step = 0; step < NSTEP; ++step) {
    // -- phase 1: w_out = [h|c] @ w_w^T + w_b   (waves 0..3, WMMA f32) --
    if (wave < 4) {
      v8f acc = {};
#pragma unroll
      for (int kk = 0; kk < 32; ++kk) {
        const int k0 = kk * 4;
        v2f a;
        if (k0 < 64) {
          a.x = s_h[arow][k0 + kbase];
          a.y = s_h[arow][k0 + kbase + 1];
        } else {
          a.x = s_c[arow][k0 - 64 + kbase];
          a.y = s_c[arow][k0 - 64 + kbase + 1];
        }
        acc = __builtin_amdgcn_wmma_f32_16x16x4_f32(
            false, a, false, Bw[kk], (short)0, acc, false, false);
      }
      const int n = wave * 16 + ccol;
#pragma unroll
      for (int i = 0; i < 8; ++i) s_wout[crow0 + i][n] = acc[i] + s_wb[n];
    }
    __syncthreads();

    // -- phase 2: v_out[w,b] = tanh(w_out[b,:] + u_out[w,b,:]) . v_w --
#pragma unroll
    for (int r = 0; r < 4; ++r) {
      const int p  = t + 256 * r;
      const int wI = p >> 4;
      const int b  = p & 15;
      const float* up = u_out + ((size_t)wI * BATCH + (b0 + b)) * E_;
      float acc = 0.f;
      for (int e = 0; e < E_; ++e)
        acc += tanhf(s_wout[b][e] + up[e]) * s_vw[e];
      s_score[wI][b] = acc;
    }
    __syncthreads();

    // -- phase 3: softmax over window axis (one thread per batch row) --
    if (t < BT) {
      float m = -3.4e38f;
      for (int w = 0; w < WIN; ++w) m = fmaxf(m, s_score[w][t]);
      float sum = 0.f;
      for (int w = 0; w < WIN; ++w) {
        const float ev = __expf(s_score[w][t] - m);
        s_score[w][t] = ev;
        sum += ev;
      }
      const float inv = 1.f / sum;
      for (int w = 0; w < WIN; ++w) s_score[w][t] *= inv;
    }
    __syncthreads();

    // -- phase 4: context[b,e] = sum_w beta[w,b] * h_enc[w,b,e] --
#pragma unroll
    for (int r = 0; r < 4; ++r) {
      const int o = t + 256 * r;
      const int b = o >> 6;
      const int e = o & 63;
      float acc = 0.f;
      for (int w = 0; w < WIN; ++w)
        acc += s_score[w][b] * h_enc[((size_t)w * BATCH + (b0 + b)) * E_ + e];
      s_ctx[b][e] = acc;
    }
    __syncthreads();

    // -- phase 5: y_tilde[b] = p_b + p_w[0]*y_prev + p_w[1:].ctx[b] --
    if (t < BT) {
      float acc = pb + s_pw[0] * y[(size_t)(b0 + t) * WIN + step];
      for (int e = 0; e < E_; ++e) acc += s_pw[1 + e] * s_ctx[t][e];
      s_ytil[t] = acc;
    }
    __syncthreads();

    // -- phase 6: gates = h @ w_hh^T + (b_ih+b_hh) + y_tilde*w_ih (WMMA) --
#pragma unroll
    for (int rep = 0; rep < 2; ++rep) {
      v8f acc = {};
#pragma unroll
      for (int kk = 0; kk < 16; ++kk) {
        v2f a;
        a.x = s_h[arow][kk * 4 + kbase];
        a.y = s_h[arow][kk * 4 + kbase + 1];
        acc = __builtin_amdgcn_wmma_f32_16x16x4_f32(
            false, a, false, Bg[rep][kk], (short)0, acc, false, false);
      }
      const int n = (wave * 2 + rep) * 16 + ccol;
#pragma unroll
      for (int i = 0; i < 8; ++i) {
        const int row = crow0 + i;
        s_gates[row][n] = acc[i] + s_bias[n] + s_ytil[row] * s_wih[n];
      }
    }
    __syncthreads();

    // -- phase 7: LSTM cell update (gate order i,f,g,o) --
#pragma unroll
    for (int r = 0; r < 4; ++r) {
      const int o = t + 256 * r;
      const int b = o >> 6;
      const int d = o & 63;
      const float gi = s_gates[b][d];
      const float gf = s_gates[b][D_ + d];
      const float gg = s_gates[b][2 * D_ + d];
      const float go = s_gates[b][3 * D_ + d];
      const float si = 1.f / (1.f + __expf(-gi));
      const float sf = 1.f / (1.f + __expf(-gf));
      const float so = 1.f / (1.f + __expf(-go));
      const float cn = sf * s_c[b][d] + si * tanhf(gg);
      s_c[b][d] = cn;
      s_h[b][d] = so * tanhf(cn);
    }
    __syncthreads();
  }

  // -- epilogue: out[b] = [h|c] @ y_w^T + y_b --
  if (t < BT) {
    float acc = y_b[0];
    for (int d = 0; d < D_; ++d)
      acc += s_yw[d] * s_h[t][d] + s_yw[D_ + d] * s_c[t][d];
    out[b0 + t] = acc;
  }
}

// ---------------------------------------------------------------------------
extern "C" void kernel_launch(void* const* d_in, const int* in_sizes, int n_in,
                              void* d_out, int out_size, void* d_ws,
                              size_t ws_size, hipStream_t stream) {
  // setup_inputs() order:
  // 0:x 1:h_encoder 2:y 3:w_ih 4:w_hh 5:b_ih 6:b_hh 7:v_w 8:w_w 9:w_b
  // 10:u_w 11:u_b 12:p_w 13:p_b 14:y_w 15:y_b
  const float* h_enc = (const float*)d_in[1];
  const float* y     = (const float*)d_in[2];
  const float* w_ih  = (const float*)d_in[3];
  const float* w_hh  = (const float*)d_in[4];
  const float* b_ih  = (const float*)d_in[5];
  const float* b_hh  = (const float*)d_in[6];
  const float* v_w   = (const float*)d_in[7];
  const float* w_w   = (const float*)d_in[8];
  const float* w_b   = (const float*)d_in[9];
  const float* u_w   = (const float*)d_in[10];
  const float* u_b   = (const float*)d_in[11];
  const float* p_w   = (const float*)d_in[12];
  const float* p_b   = (const float*)d_in[13];
  const float* y_w   = (const float*)d_in[14];
  const float* y_b   = (const float*)d_in[15];

  float* u_out = (float*)d_ws;  // needs WIN*BATCH*E_*4 = 32 MB of d_ws
  float* out   = (float*)d_out;

  u_out_kernel<<<(WIN * BATCH) / 4, 256, 0, stream>>>(h_enc, u_w, u_b, u_out);
  decoder_kernel<<<BATCH / BT, 256, 0, stream>>>(
      h_enc, y, w_ih, w_hh, b_ih, b_hh, v_w, w_w, w_b, p_w, p_b, y_w, y_b,
      u_out, out);
}